// ChebEncoder_81836306858502
// MI455X (gfx1250) — compile-verified
//
#include <hip/hip_runtime.h>
#include <hip/hip_bf16.h>

#define N_FEAT 64  // N_EMBED == N_HIDDEN == 64

typedef __attribute__((ext_vector_type(2))) float v2f;
typedef __attribute__((ext_vector_type(8))) float v8f;

// ---------------------------------------------------------------------------
// 1) per-edge degree accumulation: deg[src] += 1 for non-self-loop edges
// ---------------------------------------------------------------------------
__global__ void cheb_degree_kernel(const int* __restrict__ src,
                                   const int* __restrict__ dst,
                                   float* __restrict__ deg, int n_edges) {
    int e = blockIdx.x * blockDim.x + threadIdx.x;
    if (e < n_edges) {
        int s = src[e], d = dst[e];
        if (s != d) unsafeAtomicAdd(&deg[s], 1.0f);  // global_atomic_add_f32
    }
}

// ---------------------------------------------------------------------------
// 2) deg -> d^{-1/2} in place (0 where deg == 0)
// ---------------------------------------------------------------------------
__global__ void cheb_dinv_kernel(float* __restrict__ deg, int n_nodes) {
    int i = blockIdx.x * blockDim.x + threadIdx.x;
    if (i < n_nodes) {
        float d = deg[i];
        deg[i] = (d > 0.0f) ? __frsqrt_rn(d) : 0.0f;
    }
}

// ---------------------------------------------------------------------------
// 3) sparse propagate: T[dst] += (-dinv[src]*dinv[dst]) * X[src]
//    64 lanes per edge (one per feature); coalesced gather + f32 atomics.
// ---------------------------------------------------------------------------
__global__ void cheb_scatter_kernel(const int* __restrict__ src,
                                    const int* __restrict__ dst,
                                    const float* __restrict__ dinv,
                                    const float* __restrict__ X,
                                    float* __restrict__ T, int n_edges) {
    long long t = (long long)blockIdx.x * blockDim.x + threadIdx.x;
    int e = (int)(t >> 6);   // 64 features per edge
    int f = (int)(t & 63);
    if (e >= n_edges) return;
    int s = src[e], d = dst[e];
    if (s == d) return;                       // remove_self_loops
    float coef = -dinv[s] * dinv[d];
    if (coef == 0.0f) return;
    unsafeAtomicAdd(&T[(long long)d * N_FEAT + f],
                    coef * X[(long long)s * N_FEAT + f]);
}

// ---------------------------------------------------------------------------
// 4) fused dense part via exact-f32 WMMA:
//    out = act( X @ W0 + T @ W1 + b )
//    One block = 4 waves; wave w owns h-tile [16w,16w+16); block = 16-node tile.
//    K accumulation: 16 x V_WMMA_F32_16X16X4_F32 per operand (K=64 each).
// ---------------------------------------------------------------------------
__global__ void __launch_bounds__(128)
cheb_gemm_kernel(const float* __restrict__ X, const float* __restrict__ T,
                 const float* __restrict__ W0, const float* __restrict__ W1,
                 const float* __restrict__ bias, float* __restrict__ out,
                 int n_nodes, int do_relu) {
    const int lane = threadIdx.x & 31;
    const int wave = threadIdx.x >> 5;
    const int half = lane >> 4;   // 0: lanes 0-15, 1: lanes 16-31
    const int l15  = lane & 15;
    const int node0 = blockIdx.x * 16;
    const int h0 = wave * 16;

    // clamp row index so EXEC stays all-1s through the WMMA sequence
    int rowA = node0 + l15;
    if (rowA > n_nodes - 1) rowA = n_nodes - 1;
    const long long arow = (long long)rowA * N_FEAT;

    v8f acc = {};
    #pragma unroll
    for (int k = 0; k < N_FEAT; k += 4) {
        // A fragment: X[rowA][k + 2*half + {0,1}]  (contiguous float2)
        v2f a = *(const v2f*)(X + arow + k + 2 * half);
        // B fragment: W0[k + 2*half + {0,1}][h0 + l15]
        v2f b;
        b.x = W0[(k + 2 * half) * N_FEAT + h0 + l15];
        b.y = W0[(k + 2 * half + 1) * N_FEAT + h0 + l15];
        acc = __builtin_amdgcn_wmma_f32_16x16x4_f32(
            /*neg_a=*/false, a, /*neg_b=*/false, b,
            /*c_mod=*/(short)0, acc, /*reuse_a=*/false, /*reuse_b=*/false);
    }
    #pragma unroll
    for (int k = 0; k < N_FEAT; k += 4) {
        v2f a = *(const v2f*)(T + arow + k + 2 * half);
        v2f b;
        b.x = W1[(k + 2 * half) * N_FEAT + h0 + l15];
        b.y = W1[(k + 2 * half + 1) * N_FEAT + h0 + l15];
        acc = __builtin_amdgcn_wmma_f32_16x16x4_f32(
            false, a, false, b, (short)0, acc, false, false);
    }

    const float bv = bias[h0 + l15];
    #pragma unroll
    for (int r = 0; r < 8; ++r) {
        int row = node0 + r + 8 * half;   // C/D layout: VGPR r -> M = r (+8 for hi half)
        if (row < n_nodes) {
            float v = acc[r] + bv;
            if (do_relu) v = fmaxf(v, 0.0f);
            out[(long long)row * N_FEAT + h0 + l15] = v;
        }
    }
}

// ---------------------------------------------------------------------------
// host-side pipeline (graph-capture safe: only async ops on `stream`)
// ---------------------------------------------------------------------------
extern "C" void kernel_launch(void* const* d_in, const int* in_sizes, int n_in,
                              void* d_out, int out_size, void* d_ws, size_t ws_size,
                              hipStream_t stream) {
    const int n_edges = in_sizes[0] / 2;
    const int n_nodes = in_sizes[1] / N_FEAT;

    const int*   edge = (const int*)d_in[0];      // (2, E) row-major
    const int*   src  = edge;
    const int*   dst  = edge + n_edges;
    const float* emb  = (const float*)d_in[1];
    const float* W1_0 = (const float*)d_in[2];
    const float* W1_1 = (const float*)d_in[3];
    const float* b1   = (const float*)d_in[4];
    const float* W2_0 = (const float*)d_in[5];
    const float* W2_1 = (const float*)d_in[6];
    const float* b2   = (const float*)d_in[7];
    float* out = (float*)d_out;

    // workspace layout (floats): [deg/dinv: n_nodes][Tx: n_nodes*64][h1: n_nodes*64]
    float* deg = (float*)d_ws;
    float* Tx  = deg + n_nodes;
    float* h1  = Tx + (long long)n_nodes * N_FEAT;

    const size_t featBytes = (size_t)n_nodes * N_FEAT * sizeof(float);

    // 1) degrees -> dinv
    hipMemsetAsync(deg, 0, (size_t)n_nodes * sizeof(float), stream);
    cheb_degree_kernel<<<(n_edges + 255) / 256, 256, 0, stream>>>(src, dst, deg, n_edges);
    cheb_dinv_kernel<<<(n_nodes + 255) / 256, 256, 0, stream>>>(deg, n_nodes);

    const long long scatterThreads = (long long)n_edges * N_FEAT;
    const int scatterBlocks = (int)((scatterThreads + 255) / 256);
    const int gemmBlocks = (n_nodes + 15) / 16;

    // 2) layer 1: Tx1 = L_hat @ emb ; h1 = relu(emb@W1_0 + Tx1@W1_1 + b1)
    hipMemsetAsync(Tx, 0, featBytes, stream);
    cheb_scatter_kernel<<<scatterBlocks, 256, 0, stream>>>(src, dst, deg, emb, Tx, n_edges);
    cheb_gemm_kernel<<<gemmBlocks, 128, 0, stream>>>(emb, Tx, W1_0, W1_1, b1, h1,
                                                     n_nodes, /*relu=*/1);

    // 3) layer 2: Tx2 = L_hat @ h1 ; out = h1@W2_0 + Tx2@W2_1 + b2
    hipMemsetAsync(Tx, 0, featBytes, stream);
    cheb_scatter_kernel<<<scatterBlocks, 256, 0, stream>>>(src, dst, deg, h1, Tx, n_edges);
    cheb_gemm_kernel<<<gemmBlocks, 128, 0, stream>>>(h1, Tx, W2_0, W2_1, b2, out,
                                                     n_nodes, /*relu=*/0);
}